// InvertedIndexModel_77970836292152
// MI455X (gfx1250) — compile-verified
//
#include <hip/hip_runtime.h>
#include <stdint.h>

// Problem constants (match reference)
#define NUM_ITEMS_C 2097152
#define LWORDS      32768       // NUM_ITEMS / 64
#define WPW         64          // output words per wave in pack kernel
#define STAGE_WORDS 8           // words per TDM stage
#define NSTAGES     (WPW / STAGE_WORDS)   // 8
#define STAGE_F     (STAGE_WORDS * 64)    // 512 floats per stage
#define STAGE_B     (STAGE_F * 4)         // 2048 bytes per stage

typedef unsigned int v4u __attribute__((ext_vector_type(4)));
typedef int          v8i __attribute__((ext_vector_type(8)));
typedef int          v4i __attribute__((ext_vector_type(4)));

#if defined(__AMDGCN__)   // device pass

#if __has_builtin(__builtin_amdgcn_tensor_load_to_lds)
#define TDM_VIA_BUILTIN 1
#else
#define TDM_VIA_BUILTIN 0
#endif

#if __has_builtin(__builtin_amdgcn_s_wait_tensorcnt)
#define TWAIT(n) __builtin_amdgcn_s_wait_tensorcnt((short)(n))
#else
#define TWAIT(n) asm volatile("s_wait_tensorcnt %0" :: "i"(n) : "memory")
#endif

// Issue a 1-D TDM copy of STAGE_F f32 elements: global -> LDS.
// D# group0: count=1 | lds_addr | global_addr[56:0] | type=2.
// D# group1: data_size=4B, tensor_dim0=tile_dim0=STAGE_F, stride0=STAGE_F.
__device__ __forceinline__ void tdm_load_stage(const float* gsrc, uint32_t lds_byte_off) {
  uint64_t ga = (uint64_t)(uintptr_t)gsrc;
  v4u g0;
  g0.x = 1u;                                                // count = 1 (valid descriptor)
  g0.y = lds_byte_off;                                      // lds_addr (bytes)
  g0.z = (uint32_t)ga;                                      // global_addr[31:0]
  g0.w = (uint32_t)((ga >> 32) & 0x01FFFFFFu) | (2u << 30); // global_addr[56:32] | type=2
  v8i g1;
  g1.s0 = (int)(2u << 16);                                  // workgroup_mask=0, data_size=2 (4B)
  g1.s1 = (int)(((unsigned)STAGE_F & 0xFFFFu) << 16);       // tensor_dim0[15:0] @ bit48
  g1.s2 = (int)(((unsigned)STAGE_F >> 16) & 0xFFFFu);       // tensor_dim0[31:16]; tensor_dim1=0
  g1.s3 = (int)(((unsigned)STAGE_F & 0xFFFFu) << 16);       // tile_dim0 @ bit112
  g1.s4 = 0;                                                // tile_dim1=0, tile_dim2=0 (1-D tile)
  g1.s5 = (int)STAGE_F;                                     // tensor_dim0_stride[31:0]
  g1.s6 = 0;
  g1.s7 = 0;
  v4i z4a = {0, 0, 0, 0};
  v4i z4b = {0, 0, 0, 0};
#if TDM_VIA_BUILTIN
#if __clang_major__ >= 23
  v8i z8 = {0, 0, 0, 0, 0, 0, 0, 0};
  __builtin_amdgcn_tensor_load_to_lds(g0, g1, z4a, z4b, z8, 0);
#else
  __builtin_amdgcn_tensor_load_to_lds(g0, g1, z4a, z4b, 0);
#endif
#else
  asm volatile("tensor_load_to_lds %0, %1, %2, %3"
               :: "s"(g0), "s"(g1), "s"(z4a), "s"(z4b)
               : "memory");
#endif
}
#endif // __AMDGCN__

// Pack one 64-bit predicate word from two half-wave compares via wave32 ballots.
__device__ __forceinline__ unsigned long long pack_word(float c0, float c1, float v,
                                                        bool use_lt, bool use_gt, bool use_eq) {
  unsigned long long word = 0ull;
#if defined(__AMDGCN__)
  unsigned m0 = __builtin_amdgcn_ballot_w32(c0 < v);
  unsigned m1 = __builtin_amdgcn_ballot_w32(c1 < v);
  if (use_lt) word |= (unsigned long long)m0 | ((unsigned long long)m1 << 32);
  m0 = __builtin_amdgcn_ballot_w32(c0 > v);
  m1 = __builtin_amdgcn_ballot_w32(c1 > v);
  if (use_gt) word |= (unsigned long long)m0 | ((unsigned long long)m1 << 32);
  m0 = __builtin_amdgcn_ballot_w32(c0 == v);
  m1 = __builtin_amdgcn_ballot_w32(c1 == v);
  if (use_eq) word |= (unsigned long long)m0 | ((unsigned long long)m1 << 32);
#else
  (void)c0; (void)c1; (void)v; (void)use_lt; (void)use_gt; (void)use_eq;
#endif
  return word;
}

// Kernel 1: evaluate the 16 numeric predicates and pack into 64-bit bitmap words.
// Grid: 1024 blocks x 256 threads = 8192 waves; wave g owns predicate slot g>>9,
// words [ (g&511)*64, +64 ). Numeric data is streamed through per-wave
// double-buffered LDS stages via the Tensor Data Mover; stage reads are plain
// __shared__ accesses so they lower to ds_load_b32 (not flat), with asm memory
// fences providing compile-time ordering against the DMA writes.
__global__ __launch_bounds__(256) void iidx_pack_v4(
    const float* __restrict__ data,            // (16, NUM_ITEMS)
    const long long* __restrict__ fields,      // (16,)
    const long long* __restrict__ ops,         // (16,)
    const float* __restrict__ vals,            // (16,)
    unsigned long long* __restrict__ out_words // (16, LWORDS) in workspace
) {
#if defined(__AMDGCN__)
  __shared__ float sbuf[8 * 2 * STAGE_F];      // 8 waves x double buffer x 2KB = 32KB
#endif
  const int lane  = (int)(threadIdx.x & 31u);
  const int wid   = (int)(threadIdx.x >> 5u);
  const int gwave = (int)blockIdx.x * 8 + wid;
  const int slot  = gwave >> 9;                // predicate slot 0..15
  const int chunk = gwave & 511;
  const int w0    = chunk * WPW;

  const int   f  = (int)fields[slot];
  const float v  = vals[slot];
  const int   op = (int)ops[slot];
  const bool use_lt = (op == 1) | (op == 3);
  const bool use_gt = (op == 2) | (op == 4);
  const bool use_eq = (op == 0) | (op == 3) | (op == 4);

  const float* __restrict__ row = data + (size_t)f * NUM_ITEMS_C + (size_t)w0 * 64;
  unsigned long long* __restrict__ orow = out_words + (size_t)slot * LWORDS + (size_t)w0;

#if defined(__AMDGCN__)
  const uint32_t lds_wave = ((uint32_t)(uintptr_t)(&sbuf[0])) + (uint32_t)wid * (2u * STAGE_B);
  tdm_load_stage(row, lds_wave);                         // preload stage 0
  for (int s = 0; s < NSTAGES; ++s) {
    if (s + 1 < NSTAGES) {
      // all DS reads of the buffer we are about to overwrite have completed
      asm volatile("s_wait_dscnt 0" ::: "memory");
      tdm_load_stage(row + (size_t)(s + 1) * STAGE_F,
                     lds_wave + ((uint32_t)(s + 1) & 1u) * STAGE_B);
      TWAIT(1);                                          // stage s landed (in-order)
    } else {
      TWAIT(0);
    }
    asm volatile("" ::: "memory");                       // DMA wrote LDS: no hoisting
    const int sb = wid * (2 * STAGE_F) + (s & 1) * STAGE_F;
#pragma unroll
    for (int w = 0; w < STAGE_WORDS; ++w) {
      float c0 = sbuf[sb + w * 64 + lane];               // ds_load_b32
      float c1 = sbuf[sb + w * 64 + 32 + lane];          // ds_load_b32
      unsigned long long word = pack_word(c0, c1, v, use_lt, use_gt, use_eq);
      if (lane == 0) orow[s * STAGE_WORDS + w] = word;
    }
  }
#else
  // Host-parse-only fallback body (never executed on device): plain loads.
  for (int w = 0; w < WPW; ++w) {
    float c0 = row[w * 64 + lane];
    float c1 = row[w * 64 + 32 + lane];
    unsigned long long word = pack_word(c0, c1, v, use_lt, use_gt, use_eq);
    if (lane == 0) orow[w] = word;
  }
#endif
}

// Kernel 2: gather 16 stored bitmap rows + 16 packed numeric rows, apply
// negation/validity, AND per conjunction, OR conjunctions, mask. One thread
// per 64-bit output word; all control masks are wave-uniform scalars.
__global__ __launch_bounds__(256) void iidx_combine_v4(
    const long long* __restrict__ bitmaps,           // (512, LWORDS)
    const unsigned long long* __restrict__ np_words, // (16, LWORDS)
    const long long* __restrict__ bidx,              // (16,)
    const unsigned char* __restrict__ bvalid,        // (16,)
    const unsigned char* __restrict__ nvalid,        // (16,)
    const unsigned char* __restrict__ negm,          // (32,)
    const unsigned char* __restrict__ conj,          // (8, 32)
    const unsigned char* __restrict__ cvalid,        // (8,)
    long long* __restrict__ out,                     // (LWORDS,)
    long long num_items)
{
  const int w = (int)(blockIdx.x * blockDim.x + threadIdx.x);
  if (w >= LWORDS) return;

  const long long fw = num_items >> 6;
  const int r = (int)(num_items & 63);
  unsigned long long vm;
  if ((long long)w < fw)            vm = ~0ull;
  else if ((long long)w == fw && r) vm = (1ull << r) - 1ull;
  else                              vm = 0ull;

  unsigned long long acc[8];
#pragma unroll
  for (int c = 0; c < 8; ++c) acc[c] = ~0ull;

#pragma unroll 4
  for (int t = 0; t < 32; ++t) {
    unsigned long long rowv;
    bool valid;
    if (t < 16) {
      rowv = (unsigned long long)bitmaps[(size_t)bidx[t] * LWORDS + (size_t)w];
      valid = bvalid[t] != 0;
    } else {
      rowv = np_words[(size_t)(t - 16) * LWORDS + (size_t)w];
      valid = nvalid[t - 16] != 0;
    }
    if (negm[t]) rowv = (~rowv) & vm;   // negation first (matches reference order)
    if (!valid)  rowv = ~0ull;          // invalid rows act as all-ones
#pragma unroll
    for (int c = 0; c < 8; ++c)
      acc[c] &= (conj[c * 32 + t] ? rowv : ~0ull);
  }

  unsigned long long res = 0ull;
#pragma unroll
  for (int c = 0; c < 8; ++c)
    res |= (cvalid[c] ? acc[c] : 0ull);

  out[w] = (long long)(res & vm);
}

extern "C" void kernel_launch(void* const* d_in, const int* in_sizes, int n_in,
                              void* d_out, int out_size, void* d_ws, size_t ws_size,
                              hipStream_t stream) {
  (void)in_sizes; (void)n_in; (void)out_size; (void)ws_size;
  // Inputs in setup_inputs() dict order (reference dtypes: int64 -> long long,
  // float32 -> float, bool -> 1-byte):
  const long long*     bitmaps      = (const long long*)d_in[0];
  const float*         numeric_data = (const float*)d_in[1];
  const long long*     bidx         = (const long long*)d_in[2];
  const unsigned char* bvalid       = (const unsigned char*)d_in[3];
  const long long*     nfields      = (const long long*)d_in[4];
  const long long*     nops         = (const long long*)d_in[5];
  const float*         nvals        = (const float*)d_in[6];
  const unsigned char* nvalid       = (const unsigned char*)d_in[7];
  const unsigned char* negm         = (const unsigned char*)d_in[8];
  const unsigned char* conj         = (const unsigned char*)d_in[9];
  const unsigned char* cvalid       = (const unsigned char*)d_in[10];
  // num_items fits in 32 bits; reading the first int32 is correct for both
  // int32 and little-endian int64 storage.
  const long long num_items = (long long)((const int*)d_in[11])[0];

  unsigned long long* np_words = (unsigned long long*)d_ws;  // 16 x 32768 x 8B = 4 MiB

  iidx_pack_v4<<<dim3(1024), dim3(256), 0, stream>>>(
      numeric_data, nfields, nops, nvals, np_words);

  iidx_combine_v4<<<dim3((LWORDS + 255) / 256), dim3(256), 0, stream>>>(
      bitmaps, np_words, bidx, bvalid, nvalid, negm, conj, cvalid,
      (long long*)d_out, num_items);
}